// MAtt_38405597561125
// MI455X (gfx1250) — compile-verified
//
#include <hip/hip_runtime.h>
#include <hip/hip_bf16.h>

// ---------------------------------------------------------------------------
// Problem constants (from reference):
//   b=4, N=5, K=5, c=640, h=w=10 (hw=100), nq=75, TEMP=0.25, EPS=1e-12
// Key simplification: corr is only used via means over i or j, so
//   mean_j corr = pn^T * mean_j(qn)  and  mean_i corr = mean_i(pn)^T * qn.
// The 19.2-GFLOP einsum collapses to ~0.4 GFLOP of small GEMMs; the run time
// is dominated by the 768 MB of output stores (bandwidth bound, ~33us floor).
// ---------------------------------------------------------------------------

#define BATCH 4
#define NCLS  5
#define KSH   5
#define CH    640
#define HW    100
#define NQ    75
#define NKTOT (NCLS * KSH)          // 25
#define MROWS 500                   // N*HW
#define MPAD  512                   // padded rows for WMMA
#define QPAD  80                    // padded query count for WMMA
#define INV_TEMP 4.0f
#define EPSN 1e-12f

typedef __attribute__((ext_vector_type(2))) float v2f;
typedef __attribute__((ext_vector_type(4))) float v4f;
typedef __attribute__((ext_vector_type(8))) float v8f;

// ---------------- workspace layout (float offsets) -------------------------
#define OFF_PROTO   0ull                              // 4*5*640*100 = 1,280,000
#define OFF_PNA     1280000ull                        // 4*512*640   = 1,310,720
#define OFF_MQT     2590720ull                        // 4*640*80    =   204,800
#define OFF_MP      2795520ull                        // 4*5*640     =    12,800
#define OFF_RNP     2808320ull                        // 4*5*100     =     2,000
#define OFF_RNQ     2810320ull                        // 4*75*100    =    30,000
#define OFF_SP      2840320ull                        // 4*512*80    =   163,840
#define OFF_AP      3004160ull                        // 4*75*5*100  =   150,000
#define OFF_SQ      3154160ull                        // 150,000
#define OFF_AQ      3304160ull                        // 150,000
// total = 3,454,160 floats = 13.8 MB

// ---------------------------------------------------------------------------
// K1: proto[b,n,c,i] = mean_k fspt[b, n*K+k, c, i]
__global__ void __launch_bounds__(256) k_proto(const float* __restrict__ fspt,
                                               float* __restrict__ proto) {
    const int tid = blockIdx.x * 256 + threadIdx.x;      // 1,280,000 total
    const int i = tid % HW;
    const int c = (tid / HW) % CH;
    const int n = (tid / (HW * CH)) % NCLS;
    const int b = tid / (HW * CH * NCLS);
    float s = 0.f;
#pragma unroll
    for (int k = 0; k < KSH; ++k)
        s += fspt[(((size_t)(b * NKTOT + n * KSH + k) * CH) + c) * HW + i];
    proto[tid] = s * (1.0f / KSH);
}

// K2: rnp[b,n,i] = 1 / max(||proto[b,n,:,i]||, EPS)
__global__ void __launch_bounds__(256) k_rnp(const float* __restrict__ proto,
                                             float* __restrict__ rnp) {
    const int tid = blockIdx.x * 256 + threadIdx.x;
    if (tid >= BATCH * NCLS * HW) return;
    const int i = tid % HW;
    const int bn = tid / HW;                              // b*5+n
    const float* p = proto + ((size_t)bn * CH) * HW + i;
    float s = 0.f;
    for (int c = 0; c < CH; ++c) { float v = p[c * HW]; s += v * v; }
    rnp[tid] = 1.0f / fmaxf(sqrtf(s), EPSN);
}

// K3: rnq[b,q,j] = 1 / max(||fqry[b,q,:,j]||, EPS)
__global__ void __launch_bounds__(256) k_rnq(const float* __restrict__ fqry,
                                             float* __restrict__ rnq) {
    const int tid = blockIdx.x * 256 + threadIdx.x;
    if (tid >= BATCH * NQ * HW) return;
    const int j = tid % HW;
    const int bq = tid / HW;                              // b*75+q
    const float* p = fqry + ((size_t)bq * CH) * HW + j;
    float s = 0.f;
    for (int c = 0; c < CH; ++c) { float v = p[c * HW]; s += v * v; }
    rnq[tid] = 1.0f / fmaxf(sqrtf(s), EPSN);
}

// K4: pnA[b][r][c] = proto[b,n,c,i]*rnp[b,n,i]  (r = n*100+i, zero-pad r>=500)
__global__ void __launch_bounds__(256) k_pack_a(const float* __restrict__ proto,
                                                const float* __restrict__ rnp,
                                                float* __restrict__ pnA) {
    const int tid = blockIdx.x * 256 + threadIdx.x;      // 4*512*640
    const int c = tid % CH;
    const int r = (tid / CH) % MPAD;
    const int b = tid / (CH * MPAD);
    float v = 0.f;
    if (r < MROWS) {
        const int n = r / HW, i = r % HW;
        v = proto[((size_t)(b * NCLS + n) * CH + c) * HW + i] *
            rnp[(b * NCLS + n) * HW + i];
    }
    pnA[tid] = v;
}

// K5: mp[b,n,c] = mean_i pn[b,n,c,i]
__global__ void __launch_bounds__(256) k_mp(const float* __restrict__ proto,
                                            const float* __restrict__ rnp,
                                            float* __restrict__ mp) {
    const int tid = blockIdx.x * 256 + threadIdx.x;      // 12,800
    const int c = tid % CH;
    const int bn = tid / CH;
    const float* p = proto + ((size_t)bn * CH + c) * HW;
    const float* r = rnp + (size_t)bn * HW;
    float s = 0.f;
    for (int i = 0; i < HW; ++i) s += p[i] * r[i];
    mp[tid] = s * 0.01f;
}

// K6: mqT[b][c][qp] = mean_j qn[b,q,c,j]   (zero-pad qp>=75)
__global__ void __launch_bounds__(256) k_mqt(const float* __restrict__ fqry,
                                             const float* __restrict__ rnq,
                                             float* __restrict__ mqT) {
    const int tid = blockIdx.x * 256 + threadIdx.x;      // 4*640*80
    const int qp = tid % QPAD;
    const int c = (tid / QPAD) % CH;
    const int b = tid / (QPAD * CH);
    float v = 0.f;
    if (qp < NQ) {
        const float* p = fqry + ((size_t)(b * NQ + qp) * CH + c) * HW;
        const float* r = rnq + (size_t)(b * NQ + qp) * HW;
        float s = 0.f;
        for (int j = 0; j < HW; ++j) s += p[j] * r[j];
        v = s * 0.01f;
    }
    mqT[tid] = v;
}

// K7: WMMA GEMM  sp[b][r][q] = sum_c pnA[b][r][c] * mqT[b][c][q]
// Per b: M=512 (32 tiles), N=80 (5 tiles), K=640 (160 steps of 4).
// Uses V_WMMA_F32_16X16X4_F32; layouts per ISA 7.12.2:
//   A: VGPR0 = K{0|2 by lane half}, VGPR1 = K{1|3};  C: VGPR v -> M=v+8*(lane>=16)
__global__ void __launch_bounds__(256) k_gemm_sp(const float* __restrict__ pnA,
                                                 const float* __restrict__ mqT,
                                                 float* __restrict__ sp) {
    const int b = blockIdx.x;
    const int wave = threadIdx.x >> 5;
    const int lane = threadIdx.x & 31;
    const int lhalf = lane >> 4;
    const int l16 = lane & 15;
    const float* A = pnA + (size_t)b * MPAD * CH;
    const float* B = mqT + (size_t)b * CH * QPAD;
    float* C = sp + (size_t)b * MPAD * QPAD;

    for (int t = wave; t < 32 * 5; t += 8) {             // 20 tiles per wave
        const int mt = t / 5, nt = t % 5;
        const int m = mt * 16 + l16;
        const int ncol = nt * 16 + l16;
        const float* arow = A + (size_t)m * CH;
        v8f acc = {};
        for (int k0 = 0; k0 < CH; k0 += 4) {
            const int kb = k0 + (lhalf << 1);
            v2f av = *(const v2f*)(arow + kb);
            v2f bv;
            bv.x = B[(size_t)kb * QPAD + ncol];
            bv.y = B[(size_t)(kb + 1) * QPAD + ncol];
            acc = __builtin_amdgcn_wmma_f32_16x16x4_f32(
                false, av, false, bv, (short)0, acc, false, false);
        }
#pragma unroll
        for (int v = 0; v < 8; ++v) {
            const int row = mt * 16 + v + (lhalf << 3);
            C[(size_t)row * QPAD + ncol] = acc[v];
        }
    }
}

// K8: sq[b,q,n,j] = rnq[b,q,j] * sum_c mp[b,n,c] * fqry[b,q,c,j]
__global__ void __launch_bounds__(256) k_sq(const float* __restrict__ fqry,
                                            const float* __restrict__ mp,
                                            const float* __restrict__ rnq,
                                            float* __restrict__ sq) {
    const int tid = blockIdx.x * 256 + threadIdx.x;
    if (tid >= BATCH * NQ * HW) return;
    const int j = tid % HW;
    const int q = (tid / HW) % NQ;
    const int b = tid / (HW * NQ);
    const float* qb = fqry + ((size_t)(b * NQ + q) * CH) * HW + j;
    const float* mb = mp + (size_t)b * NCLS * CH;
    float a0 = 0, a1 = 0, a2 = 0, a3 = 0, a4 = 0;
    for (int c = 0; c < CH; ++c) {
        const float qv = qb[(size_t)c * HW];
        a0 += mb[c] * qv;
        a1 += mb[CH + c] * qv;
        a2 += mb[2 * CH + c] * qv;
        a3 += mb[3 * CH + c] * qv;
        a4 += mb[4 * CH + c] * qv;
    }
    const float rn = rnq[tid];
    float* o = sq + (size_t)(b * NQ + q) * (NCLS * HW) + j;
    o[0] = a0 * rn; o[HW] = a1 * rn; o[2 * HW] = a2 * rn;
    o[3 * HW] = a3 * rn; o[4 * HW] = a4 * rn;
}

// K9: row softmax(x/TEMP)+1 over 100 elements.
// mode 0: read sp[(b*512+n*100+i)*80+q] -> write ap[row*100+i]
// mode 1: read sq[row*100+i]            -> write aq[row*100+i]
__global__ void __launch_bounds__(128) k_softmax(const float* __restrict__ sp,
                                                 const float* __restrict__ sq,
                                                 float* __restrict__ ap,
                                                 float* __restrict__ aq,
                                                 int mode) {
    __shared__ float red[128];
    const int row = blockIdx.x;                          // 1500 rows
    const int n = row % NCLS;
    const int q = (row / NCLS) % NQ;
    const int b = row / (NCLS * NQ);
    const int tid = threadIdx.x;
    float x = -3.0e38f;
    if (tid < HW) {
        x = (mode == 0)
            ? sp[((size_t)b * MPAD + n * HW + tid) * QPAD + q]
            : sq[(size_t)row * HW + tid];
    }
    red[tid] = x; __syncthreads();
#pragma unroll
    for (int s = 64; s > 0; s >>= 1) {
        if (tid < s) red[tid] = fmaxf(red[tid], red[tid + s]);
        __syncthreads();
    }
    const float m = red[0]; __syncthreads();
    const float e = (tid < HW) ? __expf((x - m) * INV_TEMP) : 0.f;
    red[tid] = e; __syncthreads();
#pragma unroll
    for (int s = 64; s > 0; s >>= 1) {
        if (tid < s) red[tid] += red[tid + s];
        __syncthreads();
    }
    const float inv = 1.0f / red[0];
    if (tid < HW) {
        const float y = e * inv + 1.0f;
        if (mode == 0) ap[(size_t)row * HW + tid] = y;
        else           aq[(size_t)row * HW + tid] = y;
    }
}

// K10: streaming writer (bandwidth-bound part, 768 MB of NT stores).
// out1[b,q,n,c,i] = proto[b,n,c,i] * ap[b,q,n,i]
// out2[b,q,n,c,j] = fqry[b,q,c,j]  * aq[b,q,n,j]
__global__ void __launch_bounds__(256) k_write(const float* __restrict__ proto,
                                               const float* __restrict__ fqry,
                                               const float* __restrict__ ap,
                                               const float* __restrict__ aq,
                                               float* __restrict__ out) {
    const size_t HALF = 24000000ull;                     // float4 count per output
    size_t t = (size_t)blockIdx.x * 256 + threadIdx.x;   // 0..48e6
    const bool second = t >= HALF;
    if (second) t -= HALF;
    const int i = (int)(t % 25) * 4;
    size_t r = t / 25;
    const int c = (int)(r % CH); r /= CH;
    const int n = (int)(r % NCLS); r /= NCLS;
    const int q = (int)(r % NQ);
    const int b = (int)(r / NQ);

    const float* srcp = second
        ? fqry + (((size_t)(b * NQ + q) * CH + c) * HW + i)
        : proto + (((size_t)(b * NCLS + n) * CH + c) * HW + i);
    const float* wrow = (second ? aq : ap) +
        (size_t)((b * NQ + q) * NCLS + n) * HW + i;

    const v4f sv = *(const v4f*)srcp;
    const v4f wv = *(const v4f*)wrow;
    const v4f ov = sv * wv;
    v4f* dst = (v4f*)(out + (second ? 96000000ull : 0ull) +
                      ((((size_t)((b * NQ + q) * NCLS + n)) * CH + c) * HW + i));
    __builtin_nontemporal_store(ov, dst);
}

// ---------------------------------------------------------------------------
extern "C" void kernel_launch(void* const* d_in, const int* in_sizes, int n_in,
                              void* d_out, int out_size, void* d_ws, size_t ws_size,
                              hipStream_t stream) {
    const float* fspt = (const float*)d_in[0];
    const float* fqry = (const float*)d_in[1];
    float* out = (float*)d_out;
    float* ws = (float*)d_ws;

    float* proto = ws + OFF_PROTO;
    float* pnA   = ws + OFF_PNA;
    float* mqT   = ws + OFF_MQT;
    float* mp    = ws + OFF_MP;
    float* rnp   = ws + OFF_RNP;
    float* rnq   = ws + OFF_RNQ;
    float* sp    = ws + OFF_SP;
    float* ap    = ws + OFF_AP;
    float* sq    = ws + OFF_SQ;
    float* aq    = ws + OFF_AQ;

    k_proto  <<<5000,   256, 0, stream>>>(fspt, proto);
    k_rnp    <<<8,      256, 0, stream>>>(proto, rnp);
    k_rnq    <<<118,    256, 0, stream>>>(fqry, rnq);
    k_pack_a <<<5120,   256, 0, stream>>>(proto, rnp, pnA);
    k_mp     <<<50,     256, 0, stream>>>(proto, rnp, mp);
    k_mqt    <<<800,    256, 0, stream>>>(fqry, rnq, mqT);
    k_gemm_sp<<<BATCH,  256, 0, stream>>>(pnA, mqT, sp);
    k_sq     <<<118,    256, 0, stream>>>(fqry, mp, rnq, sq);
    k_softmax<<<1500,   128, 0, stream>>>(sp, sq, ap, aq, 0);
    k_softmax<<<1500,   128, 0, stream>>>(sp, sq, ap, aq, 1);
    k_write  <<<187500, 256, 0, stream>>>(proto, fqry, ap, aq, out);
}